// g_loss_19937238188075
// MI455X (gfx1250) — compile-verified
//
#include <hip/hip_runtime.h>

#define Hh 512
#define Ww 512
#define Nn 8
#define PLANE (Hh * Ww)
#define PXL (Nn * PLANE)

typedef __attribute__((ext_vector_type(2))) float v2f;
typedef __attribute__((ext_vector_type(8))) float v8f;

// ---------------------------------------------------------------------------
// Wave32 sum via V_WMMA_F32_16X16X4_F32.
// A (16x4 f32, 2 VGPRs/lane): lanes 0-15 hold K=0 (VGPR0) / K=1 (VGPR1),
// lanes 16-31 hold K=2/K=3. Put partial in VGPR0, zero in VGPR1:
//   A[m,0]=p_m, A[m,2]=p_{m+16}. B = all-ones (layout invariant).
//   D[m,n] = p_m + p_{m+16} = s_m for every n.
// C/D layout: lane n<16 VGPR j = D[j,n]=s_j ; lane n>=16 VGPR j = s_{j+8}.
// So sum of one lane's 8 D regs = s_0..7 (low half) or s_8..15 (high half);
// one shfl_xor(16) completes the full 32-lane sum.
// ---------------------------------------------------------------------------
__device__ __forceinline__ float wave_sum_wmma(float p) {
  v2f a; a.x = p;   a.y = 0.0f;
  v2f b; b.x = 1.0f; b.y = 1.0f;
  v8f c = {};
  c = __builtin_amdgcn_wmma_f32_16x16x4_f32(false, a, false, b,
                                            (short)0, c, false, false);
  float t = c[0] + c[1] + c[2] + c[3] + c[4] + c[5] + c[6] + c[7];
  t += __shfl_xor(t, 16, 32);
  return t;  // full wave sum in every lane
}

__device__ __forceinline__ float block_sum256(float p, float* sm8) {
  float w = wave_sum_wmma(p);
  const int lane = threadIdx.x & 31;
  const int wid  = threadIdx.x >> 5;
  if (lane == 0) sm8[wid] = w;
  __syncthreads();
  float s = 0.f;
  if (threadIdx.x == 0) {
    for (int k = 0; k < 8; ++k) s += sm8[k];
  }
  return s;  // valid on thread 0 only
}

// ---------------------------------------------------------------------------
// Kernel A: fused Sobel + gradient energy of v and i.
// Stages an 18x18 halo tile per source into LDS with gfx1250 async loads.
// Writes ev_raw / ei_raw (window independent) into ws.
// ---------------------------------------------------------------------------
__global__ void __launch_bounds__(256) k_energy(const float* __restrict__ v,
                                                const float* __restrict__ iv,
                                                float* __restrict__ satV,
                                                float* __restrict__ satI) {
  __shared__ float sV[18 * 18];
  __shared__ float sI[18 * 18];
  const int bid = blockIdx.x;     // 8192 blocks: 8 images x 32x32 tiles
  const int n   = bid >> 10;
  const int t   = bid & 1023;
  const int ty0 = (t >> 5) << 4;
  const int tx0 = (t & 31) << 4;
  const int tid = threadIdx.x;

  // SAME zero padding: pre-zero, then async-overwrite the in-range region.
  for (int l = tid; l < 18 * 18; l += 256) { sV[l] = 0.f; sI[l] = 0.f; }
  __syncthreads();

  const float* vb = v  + (size_t)n * PLANE;
  const float* ib = iv + (size_t)n * PLANE;
  for (int l = tid; l < 18 * 18; l += 256) {
    const int ly = l / 18, lx = l - ly * 18;
    const int gy = ty0 + ly - 1, gx = tx0 + lx - 1;
    if (gy >= 0 && gy < Hh && gx >= 0 && gx < Ww) {
      unsigned dV = (unsigned)(size_t)&sV[l];
      unsigned dI = (unsigned)(size_t)&sI[l];
      unsigned long long aV = (unsigned long long)(size_t)(vb + gy * Ww + gx);
      unsigned long long aI = (unsigned long long)(size_t)(ib + gy * Ww + gx);
      asm volatile("global_load_async_to_lds_b32 %0, %1, off"
                   :: "v"(dV), "v"(aV) : "memory");
      asm volatile("global_load_async_to_lds_b32 %0, %1, off"
                   :: "v"(dI), "v"(aI) : "memory");
    }
  }
  asm volatile("s_wait_asynccnt 0x0" ::: "memory");
  __syncthreads();

  const int tyL = tid >> 4, txL = tid & 15;
  const int c = (tyL + 1) * 18 + (txL + 1);

  float a00 = sV[c - 19], a01 = sV[c - 18], a02 = sV[c - 17];
  float a10 = sV[c - 1],                    a12 = sV[c + 1];
  float a20 = sV[c + 17], a21 = sV[c + 18], a22 = sV[c + 19];
  float gvx = (a02 + 2.f * a12 + a22) - (a00 + 2.f * a10 + a20);
  float gvy = (a20 + 2.f * a21 + a22) - (a00 + 2.f * a01 + a02);

  float b00 = sI[c - 19], b01 = sI[c - 18], b02 = sI[c - 17];
  float b10 = sI[c - 1],                    b12 = sI[c + 1];
  float b20 = sI[c + 17], b21 = sI[c + 18], b22 = sI[c + 19];
  float gix = (b02 + 2.f * b12 + b22) - (b00 + 2.f * b10 + b20);
  float giy = (b20 + 2.f * b21 + b22) - (b00 + 2.f * b01 + b02);

  const size_t o = (size_t)n * PLANE + (size_t)(ty0 + tyL) * Ww + (tx0 + txL);
  satV[o] = gvx * gvx + gvy * gvy;
  satI[o] = gix * gix + giy * giy;
}

// ---------------------------------------------------------------------------
// Kernel B1: in-place inclusive row scan (wave per row, shfl Hillis-Steele).
// Covers both energy arrays (8192 rows x 512, contiguous in ws).
// ---------------------------------------------------------------------------
__global__ void __launch_bounds__(256) k_rowscan(float* __restrict__ buf) {
  const int lane = threadIdx.x & 31;
  const int row  = blockIdx.x * 8 + (threadIdx.x >> 5);  // 1024 blocks
  float* p = buf + (size_t)row * Ww;
  float carry = 0.f;
  for (int c0 = 0; c0 < Ww; c0 += 32) {
    float x = p[c0 + lane];
#pragma unroll
    for (int d = 1; d < 32; d <<= 1) {
      float y = __shfl_up(x, d, 32);
      if (lane >= d) x += y;
    }
    x += carry;
    p[c0 + lane] = x;
    carry = __shfl(x, 31, 32);
  }
}

// ---------------------------------------------------------------------------
// Kernel B2: in-place inclusive column scan (thread per column, coalesced).
// After this, ws holds the two summed-area tables.
// ---------------------------------------------------------------------------
__global__ void __launch_bounds__(256) k_colscan(float* __restrict__ buf) {
  const int g     = blockIdx.x * 256 + threadIdx.x;  // 8192 columns total
  const int plane = g >> 9;                          // 16 planes (2 arrays x 8)
  const int x     = g & 511;
  float* p = buf + (size_t)plane * PLANE + x;
  float run = p[0];
  for (int y = 1; y < Hh; ++y) {
    run += p[(size_t)y * Ww];
    p[(size_t)y * Ww] = run;
  }
}

// ---------------------------------------------------------------------------
// Kernel C: per-pixel loss over all 5 windows (SAT lookups) + WMMA reduction.
// ---------------------------------------------------------------------------
__device__ __forceinline__ float ldz(const float* p, int y, int x) {
  return (y >= 0 && y < Hh && x >= 0 && x < Ww) ? p[y * Ww + x] : 0.f;
}

__device__ __forceinline__ float satbox(const float* S, int y, int x, int r) {
  const int y2 = min(y + r, Hh - 1), x2 = min(x + r, Ww - 1);
  const int y1 = y - r - 1, x1 = x - r - 1;
  float s = S[y2 * Ww + x2];
  if (y1 >= 0)            s -= S[y1 * Ww + x2];
  if (x1 >= 0)            s -= S[y2 * Ww + x1];
  if (y1 >= 0 && x1 >= 0) s += S[y1 * Ww + x1];
  return s;
}

__global__ void __launch_bounds__(256) k_loss(const float* __restrict__ v,
                                              const float* __restrict__ iv,
                                              const float* __restrict__ img,
                                              const float* __restrict__ satV,
                                              const float* __restrict__ satI,
                                              float* __restrict__ partials) {
  __shared__ float sm8[8];
  const int p   = blockIdx.x * 256 + threadIdx.x;
  const int n   = p >> 18;
  const int rem = p & (PLANE - 1);
  const int y   = rem >> 9;
  const int x   = rem & 511;
  const float* vp = v    + (size_t)n * PLANE;
  const float* ip = iv   + (size_t)n * PLANE;
  const float* fp = img  + (size_t)n * PLANE;
  const float* SV = satV + (size_t)n * PLANE;
  const float* SI = satI + (size_t)n * PLANE;

  __builtin_prefetch(fp + y * Ww + x, 0, 0);  // global_prefetch_b8

  float gvx, gvy, gix, giy, gfx, gfy;
  {
    float m00 = ldz(vp, y - 1, x - 1), m01 = ldz(vp, y - 1, x), m02 = ldz(vp, y - 1, x + 1);
    float m10 = ldz(vp, y,     x - 1),                           m12 = ldz(vp, y,     x + 1);
    float m20 = ldz(vp, y + 1, x - 1), m21 = ldz(vp, y + 1, x), m22 = ldz(vp, y + 1, x + 1);
    gvx = (m02 + 2.f * m12 + m22) - (m00 + 2.f * m10 + m20);
    gvy = (m20 + 2.f * m21 + m22) - (m00 + 2.f * m01 + m02);
  }
  {
    float m00 = ldz(ip, y - 1, x - 1), m01 = ldz(ip, y - 1, x), m02 = ldz(ip, y - 1, x + 1);
    float m10 = ldz(ip, y,     x - 1),                           m12 = ldz(ip, y,     x + 1);
    float m20 = ldz(ip, y + 1, x - 1), m21 = ldz(ip, y + 1, x), m22 = ldz(ip, y + 1, x + 1);
    gix = (m02 + 2.f * m12 + m22) - (m00 + 2.f * m10 + m20);
    giy = (m20 + 2.f * m21 + m22) - (m00 + 2.f * m01 + m02);
  }
  {
    float m00 = ldz(fp, y - 1, x - 1), m01 = ldz(fp, y - 1, x), m02 = ldz(fp, y - 1, x + 1);
    float m10 = ldz(fp, y,     x - 1),                           m12 = ldz(fp, y,     x + 1);
    float m20 = ldz(fp, y + 1, x - 1), m21 = ldz(fp, y + 1, x), m22 = ldz(fp, y + 1, x + 1);
    gfx = (m02 + 2.f * m12 + m22) - (m00 + 2.f * m10 + m20);
    gfy = (m20 + 2.f * m21 + m22) - (m00 + 2.f * m01 + m02);
  }

  float acc = 0.f;
  const int   rs[5]  = {1, 3, 7, 15, 30};
  const float iw2[5] = {1.f / 9.f, 1.f / 49.f, 1.f / 225.f, 1.f / 961.f, 1.f / 3721.f};
#pragma unroll
  for (int k = 0; k < 5; ++k) {
    float ev = satbox(SV, y, x, rs[k]) * iw2[k];
    float ei = satbox(SI, y, x, rs[k]) * iw2[k];
    float wt = ev / (ev + ei + 1e-8f);
    float tx = wt * gvx + (1.f - wt) * gix;
    float ty = wt * gvy + (1.f - wt) * giy;
    float dx = gfx - tx, dy = gfy - ty;
    acc += dx * dx + dy * dy;
  }

  float bs = block_sum256(acc, sm8);
  if (threadIdx.x == 0) partials[blockIdx.x] = bs;
}

// ---------------------------------------------------------------------------
// Kernel D: final reduction of 8192 block partials -> scalar mean.
// ---------------------------------------------------------------------------
__global__ void __launch_bounds__(256) k_final(const float* __restrict__ partials,
                                               float* __restrict__ out) {
  __shared__ float sm8[8];
  float a = 0.f;
  for (int i = threadIdx.x; i < 8192; i += 256) a += partials[i];
  float bs = block_sum256(a, sm8);
  if (threadIdx.x == 0) out[0] = bs * (1.0f / (float)PXL);
}

// ---------------------------------------------------------------------------
extern "C" void kernel_launch(void* const* d_in, const int* in_sizes, int n_in,
                              void* d_out, int out_size, void* d_ws, size_t ws_size,
                              hipStream_t stream) {
  const float* v   = (const float*)d_in[0];
  const float* iv  = (const float*)d_in[1];
  const float* img = (const float*)d_in[2];
  float* ws       = (float*)d_ws;
  float* satV     = ws;                 // P floats
  float* satI     = ws + (size_t)PXL;   // P floats
  float* partials = ws + 2 * (size_t)PXL;  // 8192 floats
  float* out = (float*)d_out;

  k_energy <<<8192, 256, 0, stream>>>(v, iv, satV, satI);
  k_rowscan<<<1024, 256, 0, stream>>>(ws);
  k_colscan<<<32,   256, 0, stream>>>(ws);
  k_loss   <<<8192, 256, 0, stream>>>(v, iv, img, satV, satI, partials);
  k_final  <<<1,    256, 0, stream>>>(partials, out);
}